// LIF_82927228552102
// MI455X (gfx1250) — compile-verified
//
#include <hip/hip_runtime.h>
#include <stdint.h>

// -----------------------------------------------------------------------------
// LIF neuron scan, [T=100, B=256, N=2048] fp32.
//
// Memory-bound: 420 MB total traffic, 0.5 flop/byte -> bound by 23.3 TB/s HBM
// (~18 us floor). No matmul exists in this op, so WMMA is inapplicable; the
// CDNA5 feature that matters is the async global->LDS pipeline (ASYNCcnt),
// used here as a deep per-wave prefetch FIFO to hide HBM latency across the
// serial T recurrence, plus non-temporal stores for the write-once output.
//
// Layout: thread = 4 contiguous floats (b128); block = 256 thr = 1024 floats
// per timestep; 512 blocks cover B*N = 524288. Each wave keeps DEPTH=8
// timesteps in flight via global_load_async_to_lds_b128 (saddr form:
// SGPR base = in + block offset, VGPR offset = t*BN*4 + lane*16).
// Each lane reads back exactly the 16B it staged -> no barriers; sync is
// purely s_wait_asynccnt (producer) / s_wait_dscnt (consumer-before-reuse).
// -----------------------------------------------------------------------------

typedef float v4f __attribute__((ext_vector_type(4)));

#define T_STEPS 100
#define DEPTH   8          // async pipeline depth (power of two)
#define TPB     256        // threads per block (8 wave32)
#define VEC     4          // floats per thread

__global__ __launch_bounds__(TPB) void lif_scan_async(
    const float* __restrict__ in, float* __restrict__ out, int BN)
{
    // DEPTH stages x 256 lanes x 16B = 32 KB LDS staging FIFO.
    __shared__ v4f buf[DEPTH][TPB];

    const uint32_t tid        = threadIdx.x;
    const uint32_t block_elem = (uint32_t)blockIdx.x * (TPB * VEC); // float idx within one timestep
    const unsigned long long sbase =
        (unsigned long long)(uintptr_t)(in + block_elem);           // uniform SGPR-pair base
    const uint32_t row_bytes = (uint32_t)BN * 4u;                   // stride between timesteps (2 MB)
    const uint32_t lane_lds  = tid * (uint32_t)sizeof(v4f);
    const uint32_t lds_base  = (uint32_t)(uintptr_t)(&buf[0][0]);   // low 32 bits of flat LDS addr

    // ---- prologue: launch DEPTH async loads (timesteps 0..DEPTH-1) ----------
#pragma unroll
    for (int s = 0; s < DEPTH; ++s) {
        uint32_t laddr = lds_base + (uint32_t)s * (TPB * (uint32_t)sizeof(v4f)) + lane_lds;
        uint32_t voff  = (uint32_t)s * row_bytes + lane_lds;
        asm volatile("global_load_async_to_lds_b128 %0, %1, %2"
                     :: "v"(laddr), "v"(voff), "s"(sbase) : "memory");
    }

    v4f mem = {0.0f, 0.0f, 0.0f, 0.0f};
    v4f syn = {0.0f, 0.0f, 0.0f, 0.0f};

    v4f* op = (v4f*)(out + block_elem) + tid;
    const int row_v4 = BN >> 2;
    uint32_t voff_next = (uint32_t)DEPTH * row_bytes + lane_lds;

    for (int t = 0; t < T_STEPS; ++t) {
        const int stage = t & (DEPTH - 1);

        // Outstanding loads at this point: {t .. t+DEPTH-1} (clipped at T).
        // asynccnt <= DEPTH-1  ==> load t has landed in LDS (in-order retire).
        if (t < T_STEPS - DEPTH) {
            asm volatile("s_wait_asynccnt 7" ::: "memory");
        } else if (t == T_STEPS - DEPTH) {
            asm volatile("s_wait_asynccnt 0" ::: "memory");   // drain tail
        }

        v4f x = buf[stage][tid];                              // ds_load_b128
        // Make sure the LDS read retired before we re-target this stage
        // buffer with the next async load (DS and ASYNC pipes are unordered).
        asm volatile("s_wait_dscnt 0" ::: "memory");

        if (t + DEPTH < T_STEPS) {
            uint32_t laddr = lds_base + (uint32_t)stage * (TPB * (uint32_t)sizeof(v4f)) + lane_lds;
            asm volatile("global_load_async_to_lds_b128 %0, %1, %2"
                         :: "v"(laddr), "v"(voff_next), "s"(sbase) : "memory");
            voff_next += row_bytes;
        }

        // ---- LIF update (bit-exact vs reference; 0.5*mem is exact so fma
        //      rounds identically to mul+add) ----
        v4f o;
#pragma unroll
        for (int i = 0; i < VEC; ++i) {
            float mthr = mem[i] - 1.0f;
            bool  sp   = mthr > 0.0f;
            o[i]   = sp ? 1.0f : 0.0f;
            mem[i] = sp ? 0.0f : __builtin_fmaf(0.5f, mem[i], syn[i]);
            syn[i] = x[i];
        }

        // Write-once 210 MB output stream: bypass L2 retention (NT store).
        __builtin_nontemporal_store(o, op);
        op += row_v4;
    }
}

extern "C" void kernel_launch(void* const* d_in, const int* in_sizes, int n_in,
                              void* d_out, int out_size, void* d_ws, size_t ws_size,
                              hipStream_t stream) {
    const float* in  = (const float*)d_in[0];
    float*       out = (float*)d_out;

    const int total = in_sizes[0];        // T*B*N = 52,428,800
    const int BN    = total / T_STEPS;    // 524,288 (divisible by TPB*VEC)
    const int blocks = BN / (TPB * VEC);  // 512

    lif_scan_async<<<blocks, TPB, 0, stream>>>(in, out, BN);
}